// GraphQNN__gen165_65481071399079
// MI455X (gfx1250) — compile-verified
//
#include <hip/hip_runtime.h>
#include <math.h>

#define DFEAT 128

typedef __attribute__((ext_vector_type(2))) float v2f;
typedef __attribute__((ext_vector_type(8))) float v8f;

// ---------------------------------------------------------------------------
// Native fp32 global atomic add (non-returning, device scope -> RMW at L2).
// Avoids any chance of a CAS-loop expansion from atomicAdd(float*).
// ---------------------------------------------------------------------------
__device__ __forceinline__ void atom_add_f32(float* p, float v) {
  asm volatile("global_atomic_add_f32 %0, %1, off scope:SCOPE_DEV"
               :
               : "v"(p), "v"(v)
               : "memory");
}

// ---------------------------------------------------------------------------
// Workspace zero-fill (agg + deg)
// ---------------------------------------------------------------------------
__global__ void gqnn_zero_kernel(float* __restrict__ p, size_t n) {
  size_t i = (size_t)blockIdx.x * blockDim.x + threadIdx.x;
  if (i < n) p[i] = 0.0f;
}

// ---------------------------------------------------------------------------
// Encoder: x = node_features @ W_enc^T + b_enc   (f32 WMMA 16x16x4)
// 256 threads = 8 waves; wave w owns output columns [16w,16w+16).
// Workgroup processes 8 M-tiles (128 nodes). K chunked in halves of 64.
// B-fragments (shared by all 8 M-tiles) are hoisted to registers per chunk;
// A-fragments preloaded per tile -> 16 back-to-back WMMAs per (tile,chunk).
// ---------------------------------------------------------------------------
constexpr int NT_PER_WG = 8;   // M tiles per workgroup
constexpr int KC        = 64;  // K chunk
constexpr int WT_LD     = 136; // Wt chunk stride: 2*136 % 64 == 16 -> halves on
                               // disjoint bank sets (conflict-free B reads)
constexpr int A_LD      = 68;  // A tile stride (conflict-free b64 pair reads)

__global__ __launch_bounds__(256)
void gqnn_encode_kernel(const float* __restrict__ nf, const float* __restrict__ Wenc,
                        const float* __restrict__ benc, float* __restrict__ x,
                        int N, int ntiles) {
  __shared__ float lds_wt[KC * WT_LD]; // Wt[k][n] = Wenc[n][kc+k]
  __shared__ float lds_a[16 * A_LD];   // A[row][k-kc]

  const int tid   = threadIdx.x;
  const int wave  = tid >> 5;
  const int lane  = tid & 31;
  const int lrow  = lane & 15;
  const int h     = lane >> 4;       // lane half
  const int nb    = wave * 16;       // output column base for this wave
  const int tile0 = blockIdx.x * NT_PER_WG;
  const int bcol  = nb + lrow;       // this lane's B/D column

  const float bias = benc[bcol];
  v8f acc[NT_PER_WG];
  {
    v8f vb = {bias, bias, bias, bias, bias, bias, bias, bias};
#pragma unroll
    for (int t = 0; t < NT_PER_WG; ++t) acc[t] = vb;
  }

  for (int kc = 0; kc < DFEAT; kc += KC) {
    __syncthreads(); // prior chunk's lds_wt/lds_a readers done
    // Stage W_enc^T chunk: coalesced float4 global reads, transposed LDS store.
    for (int i = tid; i < (KC / 4) * DFEAT; i += 256) {
      int n  = i >> 4;            // Wenc row (output column)
      int k4 = (i & 15) << 2;     // k offset inside chunk
      float4 w = *(const float4*)(Wenc + n * DFEAT + kc + k4);
      lds_wt[(k4 + 0) * WT_LD + n] = w.x;
      lds_wt[(k4 + 1) * WT_LD + n] = w.y;
      lds_wt[(k4 + 2) * WT_LD + n] = w.z;
      lds_wt[(k4 + 3) * WT_LD + n] = w.w;
    }
    __syncthreads(); // Wt chunk visible

    // Hoist all 16 B-fragments of this chunk (reused by all 8 M-tiles).
    // B 4x16 f32 layout: column = lane%16, K split across lane halves.
    v2f bfrag[KC / 4];
#pragma unroll
    for (int j = 0; j < KC / 4; ++j) {
      int kk = 4 * j + 2 * h;
      bfrag[j].x = lds_wt[(kk + 0) * WT_LD + bcol];
      bfrag[j].y = lds_wt[(kk + 1) * WT_LD + bcol];
    }

    for (int t = 0; t < NT_PER_WG; ++t) {
      const int  mtile  = tile0 + t;
      const bool active = (mtile < ntiles); // block-uniform
      __syncthreads(); // previous tile's lds_a readers done
      if (active) {
        int row   = tid >> 4;          // 0..15
        int c4    = (tid & 15) << 2;   // 0..60
        int gnode = mtile * 16 + row;
        float4 v  = make_float4(0.f, 0.f, 0.f, 0.f);
        if (gnode < N)
          v = *(const float4*)(nf + (size_t)gnode * DFEAT + kc + c4);
        *(float4*)&lds_a[row * A_LD + c4] = v;
      }
      __syncthreads();
      if (active) {
        // Preload all 16 A-fragments (b64 pairs), then run the WMMA chain
        // back-to-back on the accumulator.
        v2f afrag[KC / 4];
#pragma unroll
        for (int j = 0; j < KC / 4; ++j) {
          int kk = 4 * j + 2 * h;
          afrag[j].x = lds_a[lrow * A_LD + kk + 0];
          afrag[j].y = lds_a[lrow * A_LD + kk + 1];
        }
#pragma unroll
        for (int j = 0; j < KC / 4; ++j) {
          acc[t] = __builtin_amdgcn_wmma_f32_16x16x4_f32(
              false, afrag[j], false, bfrag[j], (short)0, acc[t], false, false);
        }
      }
    }
  }

  // Store: D-matrix VGPR r -> node row r + 8*h, column nb+lrow.
#pragma unroll
  for (int t = 0; t < NT_PER_WG; ++t) {
    int mtile = tile0 + t;
    if (mtile >= ntiles) break;
#pragma unroll
    for (int r = 0; r < 8; ++r) {
      int node = mtile * 16 + r + 8 * h;
      if (node < N) x[(size_t)node * DFEAT + bcol] = acc[t][r];
    }
  }
}

// ---------------------------------------------------------------------------
// Edge scatter: one wave per edge; lane l owns columns [4l, 4l+4).
// agg[dst] += x[src]; deg[dst] += 1.   (x and agg are L2-resident: 51 MB each)
// ---------------------------------------------------------------------------
__global__ __launch_bounds__(256)
void gqnn_edge_kernel(const float* __restrict__ x, const int* __restrict__ esrc,
                      const int* __restrict__ edst, float* __restrict__ agg,
                      float* __restrict__ deg, int E) {
  int e    = blockIdx.x * 8 + (threadIdx.x >> 5);
  int lane = threadIdx.x & 31;
  if (e >= E) return;
  int s = esrc[e];
  int d = edst[e];
  float4 v = ((const float4*)(x + (size_t)s * DFEAT))[lane];
  float* ar = agg + (size_t)d * DFEAT + lane * 4;
  atom_add_f32(ar + 0, v.x);
  atom_add_f32(ar + 1, v.y);
  atom_add_f32(ar + 2, v.z);
  atom_add_f32(ar + 3, v.w);
  if (lane == 0) atom_add_f32(deg + d, 1.0f);
}

// ---------------------------------------------------------------------------
// Finalize: one wave per node.
// h = x + (deg>0 ? agg/max(deg,1) : x); out = h * sigmoid(h . W_attn + b_attn)
// ---------------------------------------------------------------------------
__global__ __launch_bounds__(256)
void gqnn_final_kernel(const float* __restrict__ x, const float* __restrict__ agg,
                       const float* __restrict__ deg, const float* __restrict__ Wattn,
                       const float* __restrict__ battn, float* __restrict__ out, int N) {
  int n    = blockIdx.x * 8 + (threadIdx.x >> 5);
  int lane = threadIdx.x & 31;
  if (n >= N) return;

  float dg  = deg[n];
  float inv = 1.0f / fmaxf(dg, 1.0f);
  bool  nbr = dg > 0.0f;

  float4 xv = ((const float4*)(x   + (size_t)n * DFEAT))[lane];
  float4 av = ((const float4*)(agg + (size_t)n * DFEAT))[lane];
  float4 wv = ((const float4*)Wattn)[lane];

  float4 hv;
  hv.x = xv.x + (nbr ? av.x * inv : xv.x);
  hv.y = xv.y + (nbr ? av.y * inv : xv.y);
  hv.z = xv.z + (nbr ? av.z * inv : xv.z);
  hv.w = xv.w + (nbr ? av.w * inv : xv.w);

  float part = hv.x * wv.x + hv.y * wv.y + hv.z * wv.z + hv.w * wv.w;
#pragma unroll
  for (int off = 16; off > 0; off >>= 1) part += __shfl_xor(part, off, 32);

  float attn = 1.0f / (1.0f + __expf(-(part + battn[0])));
  float4 o = make_float4(hv.x * attn, hv.y * attn, hv.z * attn, hv.w * attn);
  ((float4*)out)[(size_t)n * (DFEAT / 4) + lane] = o;
}

// ---------------------------------------------------------------------------
// Launch
// ---------------------------------------------------------------------------
extern "C" void kernel_launch(void* const* d_in, const int* in_sizes, int n_in,
                              void* d_out, int out_size, void* d_ws, size_t ws_size,
                              hipStream_t stream) {
  (void)n_in; (void)out_size; (void)ws_size;
  const float* nf    = (const float*)d_in[0];
  const int*   esrc  = (const int*)d_in[1];
  const int*   edst  = (const int*)d_in[2];
  const float* Wenc  = (const float*)d_in[3];
  const float* benc  = (const float*)d_in[4];
  const float* Wattn = (const float*)d_in[5];
  const float* battn = (const float*)d_in[6];

  const int N = in_sizes[0] / DFEAT;
  const int E = in_sizes[1];

  float* x   = (float*)d_ws;                 // [N, 128]
  float* agg = x + (size_t)N * DFEAT;        // [N, 128]
  float* deg = agg + (size_t)N * DFEAT;      // [N]

  // zero agg + deg (contiguous)
  size_t nz = (size_t)N * DFEAT + (size_t)N;
  gqnn_zero_kernel<<<(int)((nz + 255) / 256), 256, 0, stream>>>(agg, nz);

  // encoder GEMM (WMMA)
  int ntiles = (N + 15) / 16;
  int nwg    = (ntiles + NT_PER_WG - 1) / NT_PER_WG;
  gqnn_encode_kernel<<<nwg, 256, 0, stream>>>(nf, Wenc, benc, x, N, ntiles);

  // edge gather + native fp32 atomic segment-sum
  gqnn_edge_kernel<<<(E + 7) / 8, 256, 0, stream>>>(x, esrc, edst, agg, deg, E);

  // mean / residual / attention gate
  gqnn_final_kernel<<<(N + 7) / 8, 256, 0, stream>>>(x, agg, deg, Wattn, battn,
                                                     (float*)d_out, N);
}